// InstanceSmoothnessLoss_42820823941427
// MI455X (gfx1250) — compile-verified
//
#include <hip/hip_runtime.h>
#include <hip/hip_bf16.h>
#include <stdint.h>

// InstanceSmoothnessLoss forward for MI455X (gfx1250, wave32).
//
// Roofline: N*K*C*3 ~= 0.4 GFLOP vs ~600 MB of gather traffic (mask = 64 MB,
// L2-resident on 192 MB L2; gathered rows = 512 MB from L2) -> pure
// gather-bandwidth problem. No matmul structure exists (per-pair diagonal
// distances), so WMMA would only add data movement. CDNA5 paths used:
//   * global_load_async_to_lds_b64 (ASYNCcnt) with 8 gathers in flight/wave
//   * s_wait_asynccnt draining (in-order async completion)
//   * wave32 ds_swizzle XOR butterflies for the 32-lane reductions
//   * scalar SMEM loads (s_load_b128) of the wave-uniform neighbor indices

namespace {

constexpr int C_DIM = 64;   // channels per point
constexpr int K_NN  = 8;    // neighbors per point
constexpr int WAVES = 8;    // waves per block (wave32 -> 256 threads)
constexpr int BLOCK = WAVES * 32;
constexpr int ROW_BYTES = C_DIM * (int)sizeof(float);  // 256 B

// Async gather: each lane copies 8 bytes (float2) from global memory into its
// slot of one of the wave's LDS staging buffers. One instruction == one
// ASYNCcnt increment for the wave. LDS offset = low 32 bits of the flat
// shared-aperture address (aperture base has zero low bits, ISA 10.2).
__device__ __forceinline__ void async_gather_b64(uint32_t lds_byte_off,
                                                 const void* gaddr) {
  asm volatile("global_load_async_to_lds_b64 %0, %1, off"
               :
               : "v"(lds_byte_off), "v"((uint64_t)(size_t)gaddr)
               : "memory");
}

// Full 32-lane sum via 5 XOR-swizzle stages (group-of-32 ds_swizzle pattern:
// offset = xor_mask<<10 | and_mask(0x1f)). All lanes end with the total.
__device__ __forceinline__ float wave_sum32(float v) {
  v += __int_as_float(__builtin_amdgcn_ds_swizzle(__float_as_int(v), 0x401f)); // SWAPX16
  v += __int_as_float(__builtin_amdgcn_ds_swizzle(__float_as_int(v), 0x201f)); // SWAPX8
  v += __int_as_float(__builtin_amdgcn_ds_swizzle(__float_as_int(v), 0x101f)); // SWAPX4
  v += __int_as_float(__builtin_amdgcn_ds_swizzle(__float_as_int(v), 0x081f)); // SWAPX2
  v += __int_as_float(__builtin_amdgcn_ds_swizzle(__float_as_int(v), 0x041f)); // SWAPX1
  return v;
}

}  // namespace

__global__ __launch_bounds__(BLOCK) void smooth_fwd_kernel(
    const float* __restrict__ mask,       // [N, C]
    const int*   __restrict__ nn_ind,     // [N, K]
    float*       __restrict__ out_pp,     // [N, K] per-pair distances (d_out+1)
    float*       __restrict__ block_sums, // [gridDim.x] partials for the mean
    int N) {
  __shared__ float sbuf[WAVES][K_NN][C_DIM];  // 16 KB: 8 gather bufs per wave
  __shared__ float sred[WAVES];

  const int lane = (int)(threadIdx.x & 31u);
  const int wave = __builtin_amdgcn_readfirstlane((int)(threadIdx.x >> 5));
  const int n    = (int)blockIdx.x * WAVES + wave;

  float wavesum = 0.0f;  // identical in all lanes after butterfly reductions
  float myval   = 0.0f;  // lane k keeps distance for neighbor k

  if (n < N) {
    // Self row: one coalesced 256B load per wave (float2 per lane).
    const float2 a = ((const float2*)(mask + (size_t)n * C_DIM))[lane];

    // All 8 neighbor indices up front: wave-uniform pointer -> two scalar
    // s_load_b128 (index row is 32 B, 32-B aligned).
    const int* __restrict__ idxp = nn_ind + (size_t)n * K_NN;
    const int4 iA = ((const int4*)idxp)[0];
    const int4 iB = ((const int4*)idxp)[1];
    const int j[K_NN] = {iA.x, iA.y, iA.z, iA.w, iB.x, iB.y, iB.z, iB.w};

    const uint32_t lds0 =
        (uint32_t)(size_t)(&sbuf[wave][0][0]) + (uint32_t)(lane * 8);

    // Launch all 8 async gathers: 8 outstanding 256B L2 gathers per wave.
#pragma unroll
    for (int k = 0; k < K_NN; ++k) {
      async_gather_b64(lds0 + (uint32_t)(k * ROW_BYTES),
                       mask + (size_t)j[k] * C_DIM + lane * 2);
    }

    // Drain in order: async loads return done in order, so asynccnt <= 7-k
    // guarantees buffer k has landed in LDS.
    auto body = [&](int k) {
      const float2 b = ((const float2*)(&sbuf[wave][k][0]))[lane];
      const float dx = a.x - b.x;
      const float dy = a.y - b.y;
      const float s = wave_sum32(dx * dx + dy * dy);
      const float r = (s > 0.0f) ? sqrtf(s) : 0.0f;  // safe sqrt (ref where())
      if (lane == k) myval = r;
      wavesum += r;
    };
    asm volatile("s_wait_asynccnt 7" ::: "memory"); body(0);
    asm volatile("s_wait_asynccnt 6" ::: "memory"); body(1);
    asm volatile("s_wait_asynccnt 5" ::: "memory"); body(2);
    asm volatile("s_wait_asynccnt 4" ::: "memory"); body(3);
    asm volatile("s_wait_asynccnt 3" ::: "memory"); body(4);
    asm volatile("s_wait_asynccnt 2" ::: "memory"); body(5);
    asm volatile("s_wait_asynccnt 1" ::: "memory"); body(6);
    asm volatile("s_wait_asynccnt 0" ::: "memory"); body(7);

    // 32B coalesced store of the wave's 8 distances.
    if (lane < K_NN) out_pp[(size_t)n * K_NN + lane] = myval;
  }

  if (lane == 0) sred[wave] = wavesum;
  __syncthreads();
  if (threadIdx.x == 0) {
    float t = 0.0f;
#pragma unroll
    for (int w = 0; w < WAVES; ++w) t += sred[w];
    block_sums[blockIdx.x] = t;
  }
}

// Deterministic single-block tree reduction of the per-block partials.
__global__ __launch_bounds__(256) void reduce_mean_kernel(
    const float* __restrict__ block_sums, int nb,
    float* __restrict__ out_scalar, float inv_count) {
  __shared__ float red[256];
  float t = 0.0f;
  for (int i = (int)threadIdx.x; i < nb; i += 256) t += block_sums[i];
  red[threadIdx.x] = t;
  __syncthreads();
#pragma unroll
  for (int s = 128; s > 0; s >>= 1) {
    if ((int)threadIdx.x < s) red[threadIdx.x] += red[threadIdx.x + s];
    __syncthreads();
  }
  if (threadIdx.x == 0) out_scalar[0] = red[0] * inv_count;
}

extern "C" void kernel_launch(void* const* d_in, const int* in_sizes, int n_in,
                              void* d_out, int out_size, void* d_ws,
                              size_t ws_size, hipStream_t stream) {
  const float* mask   = (const float*)d_in[0];  // [1, N, C] float32
  const int*   nn_ind = (const int*)d_in[1];    // [1, N, K] int32

  const int NK = in_sizes[1];   // N * K
  const int N  = NK / K_NN;

  float* out = (float*)d_out;   // out[0] = mean, out[1..NK] = per-pair
  float* block_sums = (float*)d_ws;

  const int nb = (N + WAVES - 1) / WAVES;

  smooth_fwd_kernel<<<nb, BLOCK, 0, stream>>>(mask, nn_ind, out + 1,
                                              block_sums, N);
  reduce_mean_kernel<<<1, 256, 0, stream>>>(block_sums, nb, out,
                                            1.0f / (float)NK);
}